// MolecFingerprintNet_90890097918354
// MI455X (gfx1250) — compile-verified
//
#include <hip/hip_runtime.h>
#include <hip/hip_bf16.h>

#define N_NODES    100000
#define N_EDGES    1600000
#define NUM_GRAPHS 512
#define D          128
#define LEVELS     3

typedef __attribute__((ext_vector_type(2))) float v2f;
typedef __attribute__((ext_vector_type(8))) float v8f;

// ---------------------------------------------------------------------------
// agg = feats  (float4-vectorized copy; the "+self" term of the aggregation)
// ---------------------------------------------------------------------------
__global__ void copy_feats_kernel(const float* __restrict__ feats,
                                  float* __restrict__ agg) {
    int i = blockIdx.x * blockDim.x + threadIdx.x;           // float4 index
    const int total = N_NODES * D / 4;                       // 3.2M
    if (i < total) {
        reinterpret_cast<float4*>(agg)[i] =
            reinterpret_cast<const float4*>(feats)[i];
    }
}

// ---------------------------------------------------------------------------
// agg[dst] += feats[src]  — one wave32 per edge, float4 per lane (512B row).
// Random rows are L2-resident (feats = 51.2MB << 192MB L2).
// ---------------------------------------------------------------------------
__global__ void edge_scatter_kernel(const float* __restrict__ feats,
                                    float* __restrict__ agg,
                                    const int* __restrict__ esrc,
                                    const int* __restrict__ edst) {
    int wave = blockIdx.x * (blockDim.x >> 5) + (threadIdx.x >> 5);
    int lane = threadIdx.x & 31;
    if (wave >= N_EDGES) return;
    int s = esrc[wave];
    int d = edst[wave];
    const float4 v = *reinterpret_cast<const float4*>(
        &feats[(size_t)s * D + lane * 4]);
    float* o = &agg[(size_t)d * D + lane * 4];
    atomicAdd(o + 0, v.x);
    atomicAdd(o + 1, v.y);
    atomicAdd(o + 2, v.z);
    atomicAdd(o + 3, v.w);
}

// ---------------------------------------------------------------------------
// Fused per-node transform for one level, one wave32 per 16-node tile:
//   h      = relu(agg @ Hw^T + Hb)          (WMMA f32 16x16x4)
//   logits = h @ Ww^T + Wb                  (WMMA f32 16x16x4, A via LDS)
//   s      = softmax(logits, axis=-1)       (register shuffles)
//   fp[graph_ids[node]] += s                (segment-reduced atomics)
// C/D layout: VGPR v, lanes 0-15 -> row v col lane; lanes 16-31 -> row v+8.
// A layout:   lanes 0-15 rows 0-15 K={k,k+1}; lanes 16-31 same rows K={k+2,k+3}.
// ---------------------------------------------------------------------------
__global__ void __launch_bounds__(32)
node_transform_kernel(const float* __restrict__ agg,
                      const float* __restrict__ Hw, const float* __restrict__ Hb,
                      const float* __restrict__ Ww, const float* __restrict__ Wb,
                      const int* __restrict__ graph_ids,
                      float* __restrict__ hout,
                      float* __restrict__ fp) {
    __shared__ float lds_h[16 * 132];          // padded: stride 132 kills bank conflicts

    const int lane  = threadIdx.x;
    const int hf    = lane >> 4;               // which 16-lane half
    const int lo    = lane & 15;
    const int node0 = blockIdx.x * 16;

    v8f acc[8];

    // ---------------- GEMM 1: h = relu(agg @ Hw^T + Hb) ----------------
    #pragma unroll
    for (int nt = 0; nt < 8; ++nt) {
        float b = Hb[nt * 16 + lo];            // bias depends only on column
        #pragma unroll
        for (int v = 0; v < 8; ++v) acc[nt][v] = b;
    }
    for (int kt = 0; kt < 32; ++kt) {
        const int k0 = kt * 4 + hf * 2;
        v2f a = *reinterpret_cast<const v2f*>(
            &agg[(size_t)(node0 + lo) * D + k0]);
        #pragma unroll
        for (int nt = 0; nt < 8; ++nt) {
            // B[k][n] = Hw[n][k] -> contiguous float2 from weight row n
            v2f b = *reinterpret_cast<const v2f*>(
                &Hw[(size_t)(nt * 16 + lo) * D + k0]);
            acc[nt] = __builtin_amdgcn_wmma_f32_16x16x4_f32(
                false, a, false, b, (short)0, acc[nt], false, false);
        }
    }

    // relu; publish h to global (next level's feats) and to LDS (A of GEMM 2)
    #pragma unroll
    for (int nt = 0; nt < 8; ++nt) {
        #pragma unroll
        for (int v = 0; v < 8; ++v) {
            float x = acc[nt][v] > 0.0f ? acc[nt][v] : 0.0f;
            int row = v + hf * 8;
            int col = nt * 16 + lo;
            hout[(size_t)(node0 + row) * D + col] = x;
            lds_h[row * 132 + col] = x;
        }
    }
    __syncthreads();

    // ---------------- GEMM 2: logits = h @ Ww^T + Wb ----------------
    #pragma unroll
    for (int nt = 0; nt < 8; ++nt) {
        float b = Wb[nt * 16 + lo];
        #pragma unroll
        for (int v = 0; v < 8; ++v) acc[nt][v] = b;
    }
    for (int kt = 0; kt < 32; ++kt) {
        const int k0 = kt * 4 + hf * 2;
        v2f a = *reinterpret_cast<const v2f*>(&lds_h[lo * 132 + k0]);
        #pragma unroll
        for (int nt = 0; nt < 8; ++nt) {
            v2f b = *reinterpret_cast<const v2f*>(
                &Ww[(size_t)(nt * 16 + lo) * D + k0]);
            acc[nt] = __builtin_amdgcn_wmma_f32_16x16x4_f32(
                false, a, false, b, (short)0, acc[nt], false, false);
        }
    }

    // ---------------- row softmax, fully in registers ----------------
    // Row (v + 8*hf) lives in VGPR v across the 16 lanes of this half.
    float rinv[8];
    #pragma unroll
    for (int v = 0; v < 8; ++v) {
        float m = acc[0][v];
        #pragma unroll
        for (int nt = 1; nt < 8; ++nt) m = fmaxf(m, acc[nt][v]);
        #pragma unroll
        for (int mask = 1; mask <= 8; mask <<= 1)
            m = fmaxf(m, __shfl_xor(m, mask, 32));   // stays within 16-lane half
        #pragma unroll
        for (int nt = 0; nt < 8; ++nt)
            acc[nt][v] = __expf(acc[nt][v] - m);
        float s = 0.0f;
        #pragma unroll
        for (int nt = 0; nt < 8; ++nt) s += acc[nt][v];
        #pragma unroll
        for (int mask = 1; mask <= 8; mask <<= 1)
            s += __shfl_xor(s, mask, 32);
        rinv[v] = 1.0f / s;
    }
    #pragma unroll
    for (int nt = 0; nt < 8; ++nt)
        #pragma unroll
        for (int v = 0; v < 8; ++v) acc[nt][v] *= rinv[v];

    // ---------------- fp[graph] += s (graph_ids sorted) ----------------
    const int g_first = graph_ids[node0];
    const int g_last  = graph_ids[node0 + 15];
    if (g_first == g_last) {
        // fast path: whole tile is one graph -> wave-reduce 16 rows first
        #pragma unroll
        for (int nt = 0; nt < 8; ++nt) {
            float s = 0.0f;
            #pragma unroll
            for (int v = 0; v < 8; ++v) s += acc[nt][v];   // rows of this half
            s += __shfl_xor(s, 16, 32);                    // add other half's rows
            if (hf == 0)
                atomicAdd(&fp[(size_t)g_first * D + nt * 16 + lo], s);
        }
    } else {
        // segment boundary tile: per-row atomics
        #pragma unroll
        for (int v = 0; v < 8; ++v) {
            int row = v + hf * 8;
            int g = graph_ids[node0 + row];
            #pragma unroll
            for (int nt = 0; nt < 8; ++nt)
                atomicAdd(&fp[(size_t)g * D + nt * 16 + lo], acc[nt][v]);
        }
    }
}

// ---------------------------------------------------------------------------
// out[g] = dot(fp[g], fc_w) + fc_b   — one wave32 per graph
// ---------------------------------------------------------------------------
__global__ void final_fc_kernel(const float* __restrict__ fp,
                                const float* __restrict__ fcw,
                                const float* __restrict__ fcb,
                                float* __restrict__ out) {
    int wave = blockIdx.x * (blockDim.x >> 5) + (threadIdx.x >> 5);
    int lane = threadIdx.x & 31;
    if (wave >= NUM_GRAPHS) return;
    const float4 a = *reinterpret_cast<const float4*>(
        &fp[(size_t)wave * D + lane * 4]);
    const float4 w = *reinterpret_cast<const float4*>(&fcw[lane * 4]);
    float s = a.x * w.x + a.y * w.y + a.z * w.z + a.w * w.w;
    #pragma unroll
    for (int mask = 16; mask >= 1; mask >>= 1) s += __shfl_xor(s, mask, 32);
    if (lane == 0) out[wave] = s + fcb[0];
}

// ---------------------------------------------------------------------------
extern "C" void kernel_launch(void* const* d_in, const int* in_sizes, int n_in,
                              void* d_out, int out_size, void* d_ws, size_t ws_size,
                              hipStream_t stream) {
    const float* x    = (const float*)d_in[0];
    const int*   esrc = (const int*)  d_in[1];
    const int*   edst = (const int*)  d_in[2];
    const int*   gids = (const int*)  d_in[3];
    const float* Hw   = (const float*)d_in[4];   // [3,128,128]
    const float* Hb   = (const float*)d_in[5];   // [3,128]
    const float* Ww   = (const float*)d_in[6];   // [3,128,128]
    const float* Wb   = (const float*)d_in[7];   // [3,128]
    const float* fcw  = (const float*)d_in[8];   // [1,128]
    const float* fcb  = (const float*)d_in[9];   // [1]
    float* out = (float*)d_out;                  // [512]

    // workspace: bufA | bufB | agg | fp   (~154 MB total)
    float* bufA = (float*)d_ws;
    float* bufB = bufA + (size_t)N_NODES * D;
    float* agg  = bufB + (size_t)N_NODES * D;
    float* fp   = agg  + (size_t)N_NODES * D;

    hipMemsetAsync(fp, 0, (size_t)NUM_GRAPHS * D * sizeof(float), stream);

    const float* feats = x;
    float* hout = bufA;
    for (int lvl = 0; lvl < LEVELS; ++lvl) {
        copy_feats_kernel<<<(N_NODES * D / 4 + 255) / 256, 256, 0, stream>>>(
            feats, agg);
        edge_scatter_kernel<<<(N_EDGES + 7) / 8, 256, 0, stream>>>(
            feats, agg, esrc, edst);
        node_transform_kernel<<<N_NODES / 16, 32, 0, stream>>>(
            agg,
            Hw + (size_t)lvl * D * D, Hb + (size_t)lvl * D,
            Ww + (size_t)lvl * D * D, Wb + (size_t)lvl * D,
            gids, hout, fp);
        feats = hout;
        hout = (hout == bufA) ? bufB : bufA;
    }
    final_fc_kernel<<<(NUM_GRAPHS + 3) / 4, 128, 0, stream>>>(fp, fcw, fcb, out);
}